// ContrastiveLearningLoss_86526411145838
// MI455X (gfx1250) — compile-verified
//
#include <hip/hip_runtime.h>

typedef float v2f __attribute__((ext_vector_type(2)));
typedef float v4f __attribute__((ext_vector_type(4)));
typedef float v8f __attribute__((ext_vector_type(8)));

#define D 256
#define ROWS_PER_BLOCK 128
#define THREADS 256
#define KCHUNK 32
#define NCHUNK (D / KCHUNK)          // 8 chunks of K
#define LDS_STRIDE (KCHUNK + 4)      // 36 floats: bank-conflict-free, 16B-aligned rows

// Stage 1: per-block partial of  sum_i (c_i * d_i + exp(d_i)),
// d_i computed as diagonal of 16x16 WMMA Gram tiles.
__global__ __launch_bounds__(THREADS)
void ContrastiveDotDiag_wmma_kernel(const float* __restrict__ za,
                                    const float* __restrict__ zb,
                                    float* __restrict__ partial,
                                    int nrows) {
    __shared__ float sa[ROWS_PER_BLOCK * LDS_STRIDE];
    __shared__ float sb[ROWS_PER_BLOCK * LDS_STRIDE];
    __shared__ float red[THREADS];

    const int t    = threadIdx.x;
    const int wave = t >> 5;         // 8 waves, 16 rows each
    const int lane = t & 31;
    const int m16  = lane & 15;
    const int half = lane >> 4;      // A/B fragment: K = vgpr + 2*half
    const int row0 = blockIdx.x * ROWS_PER_BLOCK;

    v8f acc = {};                    // 16x16 f32 accumulator (8 VGPRs)

    for (int c = 0; c < NCHUNK; ++c) {
        // ---- stage 32-column chunk of both matrices into LDS, coalesced b128 NT loads ----
        #pragma unroll
        for (int i = 0; i < (ROWS_PER_BLOCK * KCHUNK / 4) / THREADS; ++i) {  // 4 iters
            int li = i * THREADS + t;            // 0..1023 float4 index
            int r  = li >> 3;                    // 8 float4 per row-chunk
            int c4 = (li & 7) << 2;              // float column within chunk
            size_t goff = (size_t)(row0 + r) * D + (size_t)c * KCHUNK + c4;
            v4f va = __builtin_nontemporal_load((const v4f*)(za + goff));
            v4f vb = __builtin_nontemporal_load((const v4f*)(zb + goff));
            *(v4f*)(sa + r * LDS_STRIDE + c4) = va;
            *(v4f*)(sb + r * LDS_STRIDE + c4) = vb;
        }
        __syncthreads();

        // ---- 8 WMMA K-steps over this chunk (EXEC all-ones, uniform control flow) ----
        const int rbase = wave * 16;
        #pragma unroll
        for (int kb = 0; kb < KCHUNK; kb += 4) {
            int col = kb + 2 * half;
            v2f a = *(const v2f*)(sa + (rbase + m16) * LDS_STRIDE + col);
            v2f b = *(const v2f*)(sb + (rbase + m16) * LDS_STRIDE + col);
            acc = __builtin_amdgcn_wmma_f32_16x16x4_f32(
                      /*neg_a=*/false, a, /*neg_b=*/false, b,
                      /*c_mod=*/(short)0, acc, /*reuse_a=*/false, /*reuse_b=*/false);
        }
        __syncthreads();
    }

    // ---- extract the diagonal of the accumulated 16x16 tile ----
    // C layout: VGPR g, lanes 0-15 -> M=g, lanes 16-31 -> M=g+8, N = lane%16.
    // Diagonal (M==N) lives at lanes 0-7 (g=lane) and lanes 24-31 (g=lane-24).
    int g = (lane < 8) ? lane : ((lane >= 24) ? (lane - 24) : -1);
    float contrib = 0.0f;
    if (g >= 0) {
        float dv = 0.0f;
        #pragma unroll
        for (int j = 0; j < 8; ++j) dv = (j == g) ? acc[j] : dv;
        int gi = row0 + wave * 16 + g + 8 * half;     // global row index
        float coef = (gi == nrows - 1) ? (float)(nrows - 2) : (float)(nrows - 3);
        contrib = coef * dv + __expf(dv);
    }

    // ---- deterministic in-block tree reduction ----
    red[t] = contrib;
    __syncthreads();
    #pragma unroll
    for (int s = THREADS / 2; s > 0; s >>= 1) {
        if (t < s) red[t] += red[t + s];
        __syncthreads();
    }
    if (t == 0) partial[blockIdx.x] = red[0];
}

// Stage 2: deterministic reduction of per-block partials into the scalar output.
__global__ __launch_bounds__(256)
void ContrastivePartialReduce_kernel(const float* __restrict__ partial,
                                     float* __restrict__ out, int nparts) {
    __shared__ float red[256];
    int t = threadIdx.x;
    float s = 0.0f;
    for (int i = t; i < nparts; i += 256) s += partial[i];
    red[t] = s;
    __syncthreads();
    #pragma unroll
    for (int k = 128; k > 0; k >>= 1) {
        if (t < k) red[t] += red[t + k];
        __syncthreads();
    }
    if (t == 0) out[0] = red[0];
}

extern "C" void kernel_launch(void* const* d_in, const int* in_sizes, int n_in,
                              void* d_out, int out_size, void* d_ws, size_t ws_size,
                              hipStream_t stream) {
    const float* za = (const float*)d_in[0];
    const float* zb = (const float*)d_in[1];
    float* out = (float*)d_out;
    float* part = (float*)d_ws;

    int nrows   = in_sizes[0] / D;               // 65536
    int nblocks = nrows / ROWS_PER_BLOCK;        // 512 partials (2 KB of d_ws)

    ContrastiveDotDiag_wmma_kernel<<<nblocks, THREADS, 0, stream>>>(za, zb, part, nrows);
    ContrastivePartialReduce_kernel<<<1, 256, 0, stream>>>(part, out, nblocks);
}